// DCLF_4793183502637
// MI455X (gfx1250) — compile-verified
//
#include <hip/hip_runtime.h>
#include <math.h>

typedef float v2f __attribute__((ext_vector_type(2)));
typedef float v4f __attribute__((ext_vector_type(4)));
typedef float v8f __attribute__((ext_vector_type(8)));

#define WMMA_F32_16x16x4(a, b, c) \
    __builtin_amdgcn_wmma_f32_16x16x4_f32(false, (a), false, (b), (short)0, (c), false, false)

// Branchless exact-GELU: erf via Abramowitz-Stegun 7.1.26 (|err| <= 1.5e-7),
// hardware v_rcp_f32 / v_exp_f32, no EXEC divergence (unlike libm erff).
__device__ __forceinline__ float gelu_exact(float x) {
    float z  = x * 0.70710678118654752f;
    float az = fabsf(z);
    float t  = __builtin_amdgcn_rcpf(fmaf(0.3275911f, az, 1.0f));
    float p  = fmaf(1.061405429f, t, -1.453152027f);
    p = fmaf(p, t,  1.421413741f);
    p = fmaf(p, t, -0.284496736f);
    p = fmaf(p, t,  0.254829592f);
    p = p * t;
    float e  = __expf(-z * z);
    float er = copysignf(fmaf(-p, e, 1.0f), z);   // erf(z)
    return 0.5f * x * (1.0f + er);
}

// x:(16,256,256,48) -> fused [GEMM1 + gelu^2 + split + LN + dwconv3x3 + pwconv1x1 + gate + GEMM2] -> out:(16,48,256,256)
__global__ __launch_bounds__(256) void gmlp_fused_kernel(
    const float* __restrict__ x,   const float* __restrict__ W1,  const float* __restrict__ b1,
    const float* __restrict__ gam, const float* __restrict__ bet,
    const float* __restrict__ dww, const float* __restrict__ dwb,
    const float* __restrict__ pww, const float* __restrict__ pwb,
    const float* __restrict__ W2,  const float* __restrict__ b2,
    float* __restrict__ out)
{
    __shared__ float lds_n [324 * 8];   // layernormed x2 over 18x18 halo tile
    __shared__ float lds_x1[256 * 8];   // x1 gate values, interior 16x16
    __shared__ float lds_o [8 * 768];   // per-wave output transpose staging (48ch x 16px)

    const int tid  = threadIdx.x;
    const int lane = tid & 31;
    const int w    = tid >> 5;       // wave id 0..7
    const int hv   = lane >> 4;      // half-wave 0/1
    const int n    = lane & 15;      // column / pixel-in-row index

    const int blk = blockIdx.x;
    const int tx  = blk & 15;
    const int ty  = (blk >> 4) & 15;
    const int bb  = blk >> 8;
    const int gx0 = tx * 16, gy0 = ty * 16;

    const size_t xbase = (size_t)bb * (48u * 65536u);

    // ---- hoisted W1 B-fragments (12 K-steps of 4) + b1 ----
    v2f w1f[12];
#pragma unroll
    for (int k = 0; k < 12; ++k) {
        int k0 = 4 * k + 2 * hv;
        w1f[k].x = W1[(k0    ) * 16 + n];
        w1f[k].y = W1[(k0 + 1) * 16 + n];
    }
    const float b1v = b1[n];

    // ================= Phase A: GEMM1 + double GELU over 18x18 halo =================
    for (int G = w; G < 21; G += 8) {
        int pid = G * 16 + n; if (pid > 323) pid = 323;     // clamp -> duplicate rows (benign)
        int hy = pid / 18, hx = pid - 18 * hy;
        int gy = gy0 + hy - 1, gx = gx0 + hx - 1;
        int cgy = min(max(gy, 0), 255), cgx = min(max(gx, 0), 255);  // safe addr; zeroed in LN pass
        const float* xr = x + xbase + (size_t)(cgy * 256 + cgx) * 48 + 2 * hv;

        v8f acc = {};
#pragma unroll
        for (int k = 0; k < 12; ++k) {
            v2f a = *(const v2f*)(xr + 4 * k);
            acc = WMMA_F32_16x16x4(a, w1f[k], acc);
        }

#pragma unroll
        for (int i = 0; i < 8; ++i) {
            float v = gelu_exact(gelu_exact(acc[i] + b1v));
            int m = i + 8 * hv;
            int pid2 = G * 16 + m; if (pid2 > 323) pid2 = 323;
            int hy2 = pid2 / 18, hx2 = pid2 - 18 * hy2;
            if (n < 8) {   // x1 gate channel: keep only interior pixels
                if (hy2 >= 1 && hy2 <= 16 && hx2 >= 1 && hx2 <= 16)
                    lds_x1[((hy2 - 1) * 16 + (hx2 - 1)) * 8 + n] = v;
            } else {       // raw x2 channel: whole halo
                lds_n[pid2 * 8 + (n - 8)] = v;
            }
        }
    }
    __syncthreads();

    // ================= LayerNorm pass (+ zero padding outside image) =================
    {
        v4f gm0 = *(const v4f*)gam, gm1 = *(const v4f*)(gam + 4);
        v4f bt0 = *(const v4f*)bet, bt1 = *(const v4f*)(bet + 4);
        for (int pid = tid; pid < 324; pid += 256) {
            int hy = pid / 18, hx = pid - 18 * hy;
            int gy = gy0 + hy - 1, gx = gx0 + hx - 1;
            float* p = lds_n + pid * 8;
            if (gy < 0 || gy > 255 || gx < 0 || gx > 255) {
                *(v4f*)p       = (v4f){0.f, 0.f, 0.f, 0.f};
                *(v4f*)(p + 4) = (v4f){0.f, 0.f, 0.f, 0.f};
            } else {
                v4f lo = *(v4f*)p, hi = *(v4f*)(p + 4);
                float m = (lo.x + lo.y + lo.z + lo.w + hi.x + hi.y + hi.z + hi.w) * 0.125f;
                lo.x -= m; lo.y -= m; lo.z -= m; lo.w -= m;
                hi.x -= m; hi.y -= m; hi.z -= m; hi.w -= m;
                float var = (lo.x*lo.x + lo.y*lo.y + lo.z*lo.z + lo.w*lo.w
                           + hi.x*hi.x + hi.y*hi.y + hi.z*hi.z + hi.w*hi.w) * 0.125f;
                float rs = rsqrtf(var + 1e-5f);
                lo.x = lo.x * rs * gm0.x + bt0.x;  lo.y = lo.y * rs * gm0.y + bt0.y;
                lo.z = lo.z * rs * gm0.z + bt0.z;  lo.w = lo.w * rs * gm0.w + bt0.w;
                hi.x = hi.x * rs * gm1.x + bt1.x;  hi.y = hi.y * rs * gm1.y + bt1.y;
                hi.z = hi.z * rs * gm1.z + bt1.z;  hi.w = hi.w * rs * gm1.w + bt1.w;
                *(v4f*)p       = lo;
                *(v4f*)(p + 4) = hi;
            }
        }
    }
    __syncthreads();

    // ---- hoisted W2 B-fragments (3 N-tiles x 2 K-steps) + b2 ----
    const int a = 2 * hv;                 // lane's base gate channel
    float w2f[3][2][2], b2v[3];
#pragma unroll
    for (int j = 0; j < 3; ++j) {
        b2v[j] = b2[j * 16 + n];
#pragma unroll
        for (int s = 0; s < 2; ++s) {
            int k0 = 4 * s + a;
            w2f[j][s][0] = W2[(k0    ) * 48 + j * 16 + n];
            w2f[j][s][1] = W2[(k0 + 1) * 48 + j * 16 + n];
        }
    }

    // ================= Phase B: conv + gate + GEMM2 + transposed writeout =================
    for (int r = w; r < 16; r += 8) {
        const int qx = n;
        const int hy = r + 1, hx = qx + 1;

        // center n (8 channels) for pointwise conv
        const float* cp = lds_n + (hy * 18 + hx) * 8;
        v4f cn0 = *(const v4f*)cp;
        v4f cn1 = *(const v4f*)(cp + 4);

        // pointwise 1x1 for this lane's 4 channels {a, a+1, a+4, a+5}
        const int cset[4] = { a, a + 1, a + 4, a + 5 };
        float chv[4];
#pragma unroll
        for (int t = 0; t < 4; ++t) {
            int c = cset[t];
            v4f p0 = *(const v4f*)(pww + c * 8);
            v4f p1 = *(const v4f*)(pww + c * 8 + 4);
            chv[t] = pwb[c]
                   + cn0.x * p0.x + cn0.y * p0.y + cn0.z * p0.z + cn0.w * p0.w
                   + cn1.x * p1.x + cn1.y * p1.y + cn1.z * p1.z + cn1.w * p1.w;
        }

        // depthwise 3x3 (zero-padded halo already in lds_n)
        float spv[4] = { dwb[a], dwb[a + 1], dwb[a + 4], dwb[a + 5] };
#pragma unroll
        for (int j = 0; j < 9; ++j) {
            int dy = j / 3, dx = j - 3 * dy;
            const float* np = lds_n + ((hy + dy - 1) * 18 + (hx + dx - 1)) * 8;
            v2f nA = *(const v2f*)(np + a);
            v2f nB = *(const v2f*)(np + a + 4);
            spv[0] = fmaf(nA.x, dww[(a    ) * 9 + j], spv[0]);
            spv[1] = fmaf(nA.y, dww[(a + 1) * 9 + j], spv[1]);
            spv[2] = fmaf(nB.x, dww[(a + 4) * 9 + j], spv[2]);
            spv[3] = fmaf(nB.y, dww[(a + 5) * 9 + j], spv[3]);
        }

        // gate: g = x1 * (sp * ch)  -> A-fragments for GEMM2
        const float* x1p = lds_x1 + (r * 16 + qx) * 8;
        v2f x1A = *(const v2f*)(x1p + a);
        v2f x1B = *(const v2f*)(x1p + a + 4);
        v2f A0, A1;
        A0.x = x1A.x * spv[0] * chv[0];  A0.y = x1A.y * spv[1] * chv[1];
        A1.x = x1B.x * spv[2] * chv[2];  A1.y = x1B.y * spv[3] * chv[3];

        // GEMM2: (16px x 8) @ (8 x 48) as 3 tiles x 2 chained WMMAs; stage transposed in LDS
#pragma unroll
        for (int j = 0; j < 3; ++j) {
            v8f acc = {};
            v2f B0; B0.x = w2f[j][0][0]; B0.y = w2f[j][0][1];
            v2f B1; B1.x = w2f[j][1][0]; B1.y = w2f[j][1][1];
            acc = WMMA_F32_16x16x4(A0, B0, acc);
            acc = WMMA_F32_16x16x4(A1, B1, acc);
            float* op = lds_o + w * 768 + (j * 16 + n) * 16 + 8 * hv;
#pragma unroll
            for (int i = 0; i < 4; ++i) {
                v2f pr; pr.x = acc[2 * i] + b2v[j]; pr.y = acc[2 * i + 1] + b2v[j];
                *(v2f*)(op + 2 * i) = pr;
            }
        }

        // coalesced transposed writeout: out[(bb*48+c)*65536 + gy*256 + gx]
        // non-temporal: output is write-once, keep L2 for x halo reuse.
        // same-wave DS ordering guarantees the staging stores land before these loads.
#pragma unroll
        for (int it = 0; it < 6; ++it) {
            int task = it * 32 + lane;
            int c = task >> 2, quad = task & 3;
            v4f v = *(const v4f*)(lds_o + w * 768 + c * 16 + quad * 4);
            size_t off = (size_t)(bb * 48 + c) * 65536u
                       + (size_t)(gy0 + r) * 256u + (size_t)(gx0 + quad * 4);
            __builtin_nontemporal_store(v, (v4f*)(out + off));
        }
    }
}

extern "C" void kernel_launch(void* const* d_in, const int* in_sizes, int n_in,
                              void* d_out, int out_size, void* d_ws, size_t ws_size,
                              hipStream_t stream) {
    const float* x   = (const float*)d_in[0];
    const float* W1  = (const float*)d_in[1];
    const float* b1  = (const float*)d_in[2];
    const float* gam = (const float*)d_in[3];
    const float* bet = (const float*)d_in[4];
    const float* dww = (const float*)d_in[5];
    const float* dwb = (const float*)d_in[6];
    const float* pww = (const float*)d_in[7];
    const float* pwb = (const float*)d_in[8];
    const float* W2  = (const float*)d_in[9];
    const float* b2  = (const float*)d_in[10];
    float* out = (float*)d_out;

    // grid: 16 batches x (16x16) spatial tiles of 16x16 pixels
    gmlp_fused_kernel<<<dim3(4096), dim3(256), 0, stream>>>(
        x, W1, b1, gam, bet, dww, dwb, pww, pwb, W2, b2, out);
}